// CrossAttention_74972949119465
// MI455X (gfx1250) — compile-verified
//
#include <hip/hip_runtime.h>

typedef __attribute__((ext_vector_type(2))) float v2f;
typedef __attribute__((ext_vector_type(8))) float v8f;

#define DIM_ 4096
#define SEQ_ 1024
#define BSZ_ 2
#define IMG_ 512
#define KCHUNK 1024   // K elements per wave (4 waves x 1024 = K=4096)

// C[2 x 4096] = A[2 x 4096] @ W[4096 x 4096]   (fp32, WMMA 16x16x4)
// A addressed periodically: A_logical[m][k] = A[m*lda + (k & amask)]  (implements jnp.tile).
// Only C rows 0..1 are stored; lanes with m>=2 load a clamped valid A address (row m&1)
// so there is no divergence — their garbage lands in C rows 2..15 which are never stored
// (WMMA C row m depends only on A row m).
//
// Per ISA 7.12.2 (wave32, fp32):
//   A 16x4 : lanes 0-15 hold M=lane, VGPR0=K0,VGPR1=K1; lanes 16-31 hold K2,K3
//   B 4x16 : lanes 0-15 hold N=lane, VGPR0=K0,VGPR1=K1; lanes 16-31 hold K2,K3
//   C 16x16: VGPR r, lanes 0-15 -> M=r, N=lane
__global__ __launch_bounds__(128)
void gemv2_wmma_f32(const float* __restrict__ A, int lda, int amask,
                    const float* __restrict__ W, float* __restrict__ C)
{
    const int tid  = threadIdx.x;
    const int wave = tid >> 5;          // 0..3 : K-chunk owner
    const int lane = tid & 31;
    const int m    = lane & 15;         // M index (A) / N index (B)
    const int half = lane >> 4;         // selects K pair {0,1} vs {2,3}
    const int j0   = blockIdx.x << 4;   // 16-wide column strip
    const int k0   = wave * KCHUNK;

    const float* __restrict__ ap = A + (m & 1) * lda;
    const float* __restrict__ wp = W + (size_t)(k0 + 2 * half) * DIM_ + j0 + m;

    v8f acc0 = {}, acc1 = {};
    int kr = k0 + 2 * half;

#pragma unroll 8
    for (int it = 0; it < KCHUNK / 8; ++it) {
        // k-step 0
        int idx0 = kr & amask;                      // even -> {idx0, idx0+1} contiguous
        v2f a0 = *(const v2f*)(ap + idx0);
        v2f b0; b0.x = wp[0]; b0.y = wp[DIM_];
        acc0 = __builtin_amdgcn_wmma_f32_16x16x4_f32(false, a0, false, b0,
                                                     (short)0, acc0, false, false);
        // k-step 1 (independent accumulator -> shorter RAW chain)
        int idx1 = (kr + 4) & amask;
        v2f a1 = *(const v2f*)(ap + idx1);
        v2f b1; b1.x = wp[4 * DIM_]; b1.y = wp[5 * DIM_];
        acc1 = __builtin_amdgcn_wmma_f32_16x16x4_f32(false, a1, false, b1,
                                                     (short)0, acc1, false, false);
        kr += 8;
        wp += 8 * DIM_;
    }

    // cross-wave K reduction of the two live C rows
    __shared__ float red[4][2][16];
    if (lane < 16) {
        red[wave][0][m] = acc0[0] + acc1[0];   // M = 0 (batch 0)
        red[wave][1][m] = acc0[1] + acc1[1];   // M = 1 (batch 1)
    }
    __syncthreads();
    if (tid < 32) {
        const int r = tid >> 4, c = tid & 15;
        const float s = (red[0][r][c] + red[1][r][c]) + (red[2][r][c] + red[3][r][c]);
        C[r * DIM_ + j0 + c] = s;
    }
}

// out[b, q, :] = rows[b, :]  broadcast over q, float4 (b128) stores.
__global__ __launch_bounds__(256)
void bcast_rows(const float4* __restrict__ rows, float4* __restrict__ out)
{
    const size_t NP4 = DIM_ / 4;                               // 1024
    size_t i = (size_t)blockIdx.x * blockDim.x + threadIdx.x;  // over BSZ*SEQ*NP4
    size_t j = i & (NP4 - 1);
    size_t b = i >> 20;                                        // / (SEQ_ * NP4)
    out[i] = rows[b * NP4 + j];
}

extern "C" void kernel_launch(void* const* d_in, const int* in_sizes, int n_in,
                              void* d_out, int out_size, void* d_ws, size_t ws_size,
                              hipStream_t stream)
{
    // setup_inputs order: x(0) x_img(1) wq(2) wk(3) wv(4) wo(5) cos(6) sin(7) mask(8) start_pos(9)
    const float* x_img = (const float*)d_in[1];   // [2, 512]
    const float* wv    = (const float*)d_in[4];   // [4096, 4096]
    const float* wo    = (const float*)d_in[5];   // [4096, 4096]

    float* vbase = (float*)d_ws;          // [2, 4096] = tile8(x_img) @ wv
    float* frow  = vbase + BSZ_ * DIM_;   // [2, 4096] = vbase @ wo

    dim3 gg(DIM_ / 16), gb(128);
    // v_base[b,:] = tile8(x_img[b,:]) @ wv   (A addressed mod 512)
    gemv2_wmma_f32<<<gg, gb, 0, stream>>>(x_img, IMG_, IMG_ - 1, wv, vbase);
    // final_row[b,:] = v_base[b,:] @ wo
    gemv2_wmma_f32<<<gg, gb, 0, stream>>>(vbase, DIM_, DIM_ - 1, wo, frow);
    // out[b,q,:] = final_row[b,:]
    const int total4 = BSZ_ * SEQ_ * (DIM_ / 4);
    bcast_rows<<<total4 / 256, 256, 0, stream>>>((const float4*)frow, (float4*)d_out);
}